// SpikeLayer_33122787787178
// MI455X (gfx1250) — compile-verified
//
#include <hip/hip_runtime.h>
#include <stdint.h>

// LIF recurrence over T for x:[B, T, C, H, W] fp32. Memory-bound streaming
// kernel: async global->LDS b128 loads (ASYNCcnt path) with staggered
// s_wait_asynccnt, ds_load_b128 reads, VALU recurrence in registers,
// nontemporal b128 stores.

typedef float v4f __attribute__((ext_vector_type(4)));

#define T_FULL 8
#define CHW    262144      // 256*32*32 (power of two)
#define CHW_SHIFT 18
#define VTH    0.5f
#define DECAY  0.25f

__device__ __forceinline__ void lif_step(const v4f* lp, int idx, v4f& mem,
                                         float* outp)
{
    v4f xi = lp[idx];                              // ds_load_b128
    v4f sp;
    #pragma unroll
    for (int c = 0; c < 4; ++c) {
        float m = fmaf(mem[c], DECAY, xi[c]);      // mem = mem*decay + x
        bool s  = (m - VTH) > 0.0f;                // Heaviside forward
        sp[c]   = s ? 1.0f : 0.0f;
        mem[c]  = s ? 0.0f : m;                    // hard reset where spiked
    }
    __builtin_nontemporal_store(sp, (v4f*)outp);   // NT b128 store
}

__global__ __launch_bounds__(256) void lif_spike_kernel(
    const float* __restrict__ x, float* __restrict__ out,
    int now_T, long long n_thread)
{
    // [t][tid] tiles of 16B each: 8 * 256 * 16B = 32 KiB
    __shared__ float lbuf[T_FULL * 256 * 4];

    const int tid = threadIdx.x;
    const long long gid = (long long)blockIdx.x * 256 + tid;
    if (gid >= n_thread) return;

    const long long sp4 = gid * 4;                 // flat spatial float index
    const int b   = (int)(sp4 >> CHW_SHIFT);       // batch
    const int chw = (int)(sp4 & (CHW - 1));        // position within plane

    // byte offset of x[b, 0, chw]; max ~529 MB, fits signed 32-bit VGPR offset
    const uint32_t in_byte = (uint32_t)(((uint32_t)b * (T_FULL * CHW) + (uint32_t)chw) * 4u);

    // LDS byte address of this lane's slot for t=0 (low 32 bits of a generic
    // pointer to a __shared__ object are the LDS offset on gfx1250)
    const uint32_t lbase = (uint32_t)(uintptr_t)(&lbuf[0]) + (uint32_t)tid * 16u;

    // Issue all T async b128 loads up-front: 8 independent in-flight requests
    // per lane -> deep MLP against HBM latency. GVS form: SGPR64 base + VGPR
    // 32-bit byte offset. Tracked by ASYNCcnt; loads complete in order.
    #pragma unroll
    for (int t = 0; t < T_FULL; ++t) {
        uint32_t la = lbase + (uint32_t)t * (256u * 16u);
        uint32_t va = in_byte + (uint32_t)t * (CHW * 4u);
        asm volatile("global_load_async_to_lds_b128 %0, %1, %2"
                     :: "v"(la), "v"(va), "s"(x)
                     : "memory");
    }

    // Each lane only reads back the LDS words its own async loads wrote, so no
    // barrier is needed, just ASYNCcnt waits.
    const v4f* lp = (const v4f*)lbuf;
    float* outb = out + ((long long)b * now_T) * (long long)CHW + chw;

    v4f mem = {0.0f, 0.0f, 0.0f, 0.0f};

    if (now_T == T_FULL) {
        // Common path: staggered waits. Async loads retire in order, so tile t
        // is ready once ASYNCcnt <= 7-t; stores start as soon as tile 0 lands,
        // overlapping the write stream with the remaining in-flight reads.
        #define LIF_T(t, wcnt)                                              \
            asm volatile("s_wait_asynccnt " #wcnt ::: "memory");            \
            lif_step(lp, (t) * 256 + tid, mem, outb + (long long)(t) * CHW);
        LIF_T(0, 7)
        LIF_T(1, 6)
        LIF_T(2, 5)
        LIF_T(3, 4)
        LIF_T(4, 3)
        LIF_T(5, 2)
        LIF_T(6, 1)
        LIF_T(7, 0)
        #undef LIF_T
    } else {
        // Generic fallback for any now_T in [1, 8]
        asm volatile("s_wait_asynccnt 0" ::: "memory");
        for (int t = 0; t < now_T; ++t)
            lif_step(lp, t * 256 + tid, mem, outb + (long long)t * CHW);
    }
}

extern "C" void kernel_launch(void* const* d_in, const int* in_sizes, int n_in,
                              void* d_out, int out_size, void* d_ws, size_t ws_size,
                              hipStream_t stream) {
    (void)n_in; (void)d_ws; (void)ws_size;
    const float* x = (const float*)d_in[0];
    float* out = (float*)d_out;

    const long long total_in = (long long)in_sizes[0];      // B * 8 * C*H*W
    const long long spatial  = total_in / T_FULL;           // B * C*H*W
    const int now_T          = (int)((long long)out_size / spatial);
    const long long n_thread = spatial / 4;                 // 4 floats / thread
    const int blocks         = (int)((n_thread + 255) / 256);

    lif_spike_kernel<<<blocks, 256, 0, stream>>>(x, out, now_T, n_thread);
}